// ScoreAggregation_27960237097208
// MI455X (gfx1250) — compile-verified
//
#include <hip/hip_runtime.h>
#include <hip/hip_bf16.h>

#define N_NODES 4096
#define N_TYPES 3
#define N_HEADS 4
#define N_EDGES 65536
#define NEG_SLOPE 0.2f

// nibble-packed type mask: entry (i,j) -> 4 bits (bits 0..2 = type flags)
// words per row = 4096/8 = 512 ; total words = 4096*512 = 2,097,152 (8 MB)
#define WORDS_PER_ROW 512
#define MASK_WORDS (N_NODES * WORDS_PER_ROW)

typedef __attribute__((ext_vector_type(2))) float v2f;
typedef __attribute__((ext_vector_type(8))) float v8f;

// ---------------- kernel 1: zero the mask ----------------
__global__ __launch_bounds__(256) void zero_mask_kernel(uint4* p, int n4) {
    int i = blockIdx.x * blockDim.x + threadIdx.x;
    if (i < n4) p[i] = make_uint4(0u, 0u, 0u, 0u);
}

// ---------------- kernel 2: constants (single wave32, uses WMMA) ----------------
// cb layout (floats): [0]=S_total ; [4..7]=w_src[h] ; [8..11]=w_tgt[h] ;
//                     [16 + t*4 + h] = c[t][h]
__global__ __launch_bounds__(32) void consts_kernel(const float* __restrict__ scores,
                                                    const float* __restrict__ emb,
                                                    const float* __restrict__ attw,
                                                    float* __restrict__ cb) {
    int lane = threadIdx.x;           // 0..31, full wave -> EXEC all ones
    int g = lane >> 4;                // lane half (selects K pair)
    int m = lane & 15;                // M index (A) / N index (B)

    // --- S_total = sum(scores) via wave reduction ---
    float acc = 0.f;
    for (int i = lane; i < N_NODES; i += 32) acc += scores[i];
    for (int off = 16; off > 0; off >>= 1) acc += __shfl_down(acc, off);

    // --- c[t][h] = sum_d emb[t][d] * attw[h][1+d] via V_WMMA_F32_16X16X4_F32 chain ---
    // A (16x4 f32): lane L, vgpr v holds A[L&15][ (L>>4)*2 + v ]   (ISA 7.12.2)
    // B (4x16 f32): lane L, vgpr v holds B[ (L>>4)*2 + v ][ L&15 ] (symmetric layout)
    v8f d = {0.f, 0.f, 0.f, 0.f, 0.f, 0.f, 0.f, 0.f};
    #pragma unroll
    for (int kk = 0; kk < 4; ++kk) {
        int k0 = g * 2 + 0, k1 = g * 2 + 1;
        v2f a, b;
        a.x = (m < N_TYPES) ? emb[m * 16 + kk * 4 + k0] : 0.f;
        a.y = (m < N_TYPES) ? emb[m * 16 + kk * 4 + k1] : 0.f;
        b.x = (m < N_HEADS) ? attw[m * 18 + 1 + kk * 4 + k0] : 0.f;
        b.y = (m < N_HEADS) ? attw[m * 18 + 1 + kk * 4 + k1] : 0.f;
        d = __builtin_amdgcn_wmma_f32_16x16x4_f32(false, a, false, b,
                                                  (short)0, d, false, false);
    }
    // D layout: vgpr r: lanes0-15 -> M=r,N=lane ; so c[t][h] sits in d[t], lane h
    if (lane < N_HEADS) {
        cb[16 + 0 * 4 + lane] = d[0];
        cb[16 + 1 * 4 + lane] = d[1];
        cb[16 + 2 * 4 + lane] = d[2];
    }
    if (lane == 0) {
        cb[0] = acc;
        #pragma unroll
        for (int h = 0; h < N_HEADS; ++h) {
            cb[4 + h] = attw[h * 18 + 0];       // w_src
            cb[8 + h] = attw[h * 18 + 17];      // w_tgt
        }
    }
}

// ---------------- kernel 3: scatter edges into nibble mask ----------------
__global__ __launch_bounds__(256) void scatter_kernel(const long long* __restrict__ el,
                                                      unsigned* __restrict__ mask) {
    int idx = blockIdx.x * blockDim.x + threadIdx.x;
    if (idx >= N_TYPES * N_EDGES) return;
    int t = idx >> 16;                 // type
    int e = idx & (N_EDGES - 1);
    long long base = (long long)t * 2 * N_EDGES;
    unsigned src = (unsigned)el[base + e];
    unsigned dst = (unsigned)el[base + N_EDGES + e];
    unsigned a1 = src * (unsigned)N_NODES + dst;
    unsigned a2 = dst * (unsigned)N_NODES + src;
    atomicOr(&mask[a1 >> 3], 1u << (((a1 & 7u) << 2) + t));
    atomicOr(&mask[a2 >> 3], 1u << (((a2 & 7u) << 2) + t));
}

// ---------------- kernel 4: per-row softmax correction ----------------
__global__ __launch_bounds__(256) void row_kernel(const unsigned* __restrict__ mask,
                                                  const float* __restrict__ scores,
                                                  const float* __restrict__ cb,
                                                  float* __restrict__ out) {
    int i = blockIdx.x;
    int tid = threadIdx.x;
    float s_i = scores[i];
    float S_total = cb[0];
    float wsrc[N_HEADS], wtgt[N_HEADS], c[N_TYPES][N_HEADS];
    #pragma unroll
    for (int h = 0; h < N_HEADS; ++h) { wsrc[h] = cb[4 + h]; wtgt[h] = cb[8 + h]; }
    #pragma unroll
    for (int t = 0; t < N_TYPES; ++t)
        #pragma unroll
        for (int h = 0; h < N_HEADS; ++h) c[t][h] = cb[16 + t * 4 + h];

    float dden[N_HEADS] = {0.f, 0.f, 0.f, 0.f};
    float dnum[N_HEADS] = {0.f, 0.f, 0.f, 0.f};

    const unsigned* row = mask + (size_t)i * WORDS_PER_ROW;
    for (int w = tid; w < WORDS_PER_ROW; w += 256) {
        unsigned bits = row[w];
        if (!bits) continue;
        #pragma unroll
        for (int nib = 0; nib < 8; ++nib) {
            unsigned mm = (bits >> (nib * 4)) & 7u;
            if (!mm) continue;
            int j = w * 8 + nib;
            float s_j = scores[j];
            float pc = (float)__popc(mm);
            float base = wsrc[0]; (void)base;
            #pragma unroll
            for (int h = 0; h < N_HEADS; ++h) {
                float cs = 0.f;
                if (mm & 1u) cs += c[0][h];
                if (mm & 2u) cs += c[1][h];
                if (mm & 4u) cs += c[2][h];
                float l = pc * (wsrc[h] * s_i + wtgt[h] * s_j) + cs;
                l = (l >= 0.f) ? l : NEG_SLOPE * l;
                float em1 = expf(l) - 1.f;
                dden[h] += em1;
                dnum[h] += em1 * s_j;
            }
        }
    }

    __shared__ float red[2 * N_HEADS][256];
    #pragma unroll
    for (int h = 0; h < N_HEADS; ++h) {
        red[h][tid] = dden[h];
        red[N_HEADS + h][tid] = dnum[h];
    }
    __syncthreads();
    for (int s = 128; s > 0; s >>= 1) {
        if (tid < s) {
            #pragma unroll
            for (int k = 0; k < 2 * N_HEADS; ++k) red[k][tid] += red[k][tid + s];
        }
        __syncthreads();
    }
    if (tid == 0) {
        float o = 0.f;
        #pragma unroll
        for (int h = 0; h < N_HEADS; ++h)
            o += (S_total + red[N_HEADS + h][0]) / ((float)N_NODES + red[h][0]);
        out[i] = o * (1.f / (float)N_HEADS);
    }
}

extern "C" void kernel_launch(void* const* d_in, const int* in_sizes, int n_in,
                              void* d_out, int out_size, void* d_ws, size_t ws_size,
                              hipStream_t stream) {
    const float*     scores = (const float*)d_in[0];      // (4096,1) f32
    const float*     emb    = (const float*)d_in[1];      // (3,16)   f32
    const float*     attw   = (const float*)d_in[2];      // (4,18,1) f32
    const long long* el     = (const long long*)d_in[3];  // (3,2,65536) i64
    float* out = (float*)d_out;                           // 4096 f32

    float*    cb   = (float*)d_ws;                        // 256 B of constants
    unsigned* mask = (unsigned*)((char*)d_ws + 256);      // 8 MB nibble mask

    // 1) zero 8MB mask (uint4 stores: 524288 x 16B)
    zero_mask_kernel<<<(MASK_WORDS / 4 + 255) / 256, 256, 0, stream>>>((uint4*)mask,
                                                                       MASK_WORDS / 4);
    // 2) device-side constants: S_total, w_src/w_tgt, c = emb @ w_mid via WMMA
    consts_kernel<<<1, 32, 0, stream>>>(scores, emb, attw, cb);
    // 3) scatter edges (dedup via atomicOr — matches .set(1.0) semantics)
    scatter_kernel<<<(N_TYPES * N_EDGES + 255) / 256, 256, 0, stream>>>(el, mask);
    // 4) one block per row: scan mask, accumulate softmax corrections
    row_kernel<<<N_NODES, 256, 0, stream>>>(mask, scores, cb, out);
}